// AllGeomLoss_21827023799004
// MI455X (gfx1250) — compile-verified
//
#include <hip/hip_runtime.h>

typedef float v2f __attribute__((ext_vector_type(2)));
typedef float v8f __attribute__((ext_vector_type(8)));

#define B_N 4096
#define D_N 64
#define K_NB 25

// Workspace layout (float offsets into d_ws)
#define WS_RECON 0      // recon sum-of-squares accumulator
#define WS_TRACE 1
#define WS_FROB  2
#define WS_LMAX  3
#define WS_TSA   4      // sum over points of (2 - 2*(u.v)^2)
#define WS_MEAN  8      // 64 column means of latent
#define WS_COV   80     // 64x64 covariance
#define WS_SQ    4224   // 4096 squared row norms of raw
#define WS_NBR   8320   // int region: 4096*25 neighbor indices
#define WS_D2    114688 // 4096*4096 pairwise squared distances (64 MB)

// ---------- init: zero scalar accumulators ----------
__global__ void k_init(float* fs) {
    if (threadIdx.x < 8) fs[threadIdx.x] = 0.f;
}

// ---------- recon = mean((outputs-targets)^2) ----------
__global__ void k_recon(const float* __restrict__ o, const float* __restrict__ t,
                        float* fs) {
    __shared__ float red[256];
    int tid = threadIdx.x;
    float s = 0.f;
    for (int i = blockIdx.x * 256 + tid; i < B_N * D_N; i += gridDim.x * 256) {
        float d = o[i] - t[i];
        s += d * d;
    }
    red[tid] = s; __syncthreads();
    for (int w = 128; w > 0; w >>= 1) {
        if (tid < w) red[tid] += red[tid + w];
        __syncthreads();
    }
    if (tid == 0) atomicAdd(&fs[WS_RECON], red[0]);
}

// ---------- column means of latent ----------
__global__ void k_colmean(const float* __restrict__ z, float* fs) {
    __shared__ float red[256];
    int j = blockIdx.x, tid = threadIdx.x;
    float s = 0.f;
    for (int b = tid; b < B_N; b += 256) s += z[b * D_N + j];
    red[tid] = s; __syncthreads();
    for (int w = 128; w > 0; w >>= 1) {
        if (tid < w) red[tid] += red[tid + w];
        __syncthreads();
    }
    if (tid == 0) fs[WS_MEAN + j] = red[0] / (float)B_N;
}

// ---------- covariance row j of centered latent ----------
__global__ void k_cov(const float* __restrict__ z, float* fs) {
    __shared__ float mean[64];
    __shared__ float crow[64];
    int j = blockIdx.x, tid = threadIdx.x;
    if (tid < 64) { mean[tid] = fs[WS_MEAN + tid]; crow[tid] = 0.f; }
    __syncthreads();
    float acc[64];
#pragma unroll
    for (int k = 0; k < 64; ++k) acc[k] = 0.f;
    float mj = mean[j];
    for (int b = tid; b < B_N; b += 256) {
        float zj = z[b * D_N + j] - mj;
#pragma unroll
        for (int k = 0; k < 64; ++k) acc[k] += zj * (z[b * D_N + k] - mean[k]);
    }
#pragma unroll
    for (int k = 0; k < 64; ++k) atomicAdd(&crow[k], acc[k]);
    __syncthreads();
    if (tid < 64) fs[WS_COV + j * 64 + tid] = crow[tid] / ((float)(B_N - 1) + 1e-8f);
}

// ---------- trace, Frobenius^2, lambda_max (power iteration), 1 block x 64 thr ----------
__global__ void k_eig(float* fs) {
    __shared__ float C[64 * 64];
    __shared__ float v[64], w[64], red[64];
    int t = threadIdx.x;
    for (int r = 0; r < 64; ++r) C[r * 64 + t] = fs[WS_COV + r * 64 + t];
    __syncthreads();
    float fr = 0.f;
#pragma unroll
    for (int k = 0; k < 64; ++k) { float c = C[t * 64 + k]; fr += c * c; }
    red[t] = fr; __syncthreads();
    for (int s = 32; s > 0; s >>= 1) { if (t < s) red[t] += red[t + s]; __syncthreads(); }
    float frob = red[0]; __syncthreads();
    red[t] = C[t * 64 + t]; __syncthreads();
    for (int s = 32; s > 0; s >>= 1) { if (t < s) red[t] += red[t + s]; __syncthreads(); }
    float trace = red[0]; __syncthreads();
    v[t] = 1.f + 0.01f * (float)t; __syncthreads();
    for (int it = 0; it < 200; ++it) {
        float s = 0.f;
#pragma unroll
        for (int k = 0; k < 64; ++k) s += C[t * 64 + k] * v[k];
        w[t] = s; red[t] = s * s; __syncthreads();
        for (int st = 32; st > 0; st >>= 1) { if (t < st) red[t] += red[t + st]; __syncthreads(); }
        float nr = rsqrtf(red[0] + 1e-30f);
        __syncthreads();
        v[t] = w[t] * nr; __syncthreads();
    }
    float s = 0.f;
#pragma unroll
    for (int k = 0; k < 64; ++k) s += C[t * 64 + k] * v[k];
    red[t] = v[t] * s; __syncthreads();
    for (int st = 32; st > 0; st >>= 1) { if (t < st) red[t] += red[t + st]; __syncthreads(); }
    if (t == 0) { fs[WS_TRACE] = trace; fs[WS_FROB] = frob; fs[WS_LMAX] = red[0]; }
}

// ---------- squared row norms of raw ----------
__global__ void k_sq(const float* __restrict__ raw, float* fs) {
    int i = blockIdx.x * 256 + threadIdx.x;
    if (i < B_N) {
        float s = 0.f;
#pragma unroll
        for (int k = 0; k < 64; ++k) { float x = raw[i * 64 + k]; s += x * x; }
        fs[WS_SQ + i] = s;
    }
}

// ---------- Gram via V_WMMA_F32_16X16X4_F32; d2 = sq_i + sq_j - 2*G ----------
// 256 threads = 8 waves, each wave owns one 16x16 output tile.
__global__ void k_gram(const float* __restrict__ raw, float* __restrict__ fs) {
    float* d2 = fs + WS_D2;
    const float* sq = fs + WS_SQ;
    int wave = threadIdx.x >> 5;
    int lane = threadIdx.x & 31;
    int tile = blockIdx.x * 8 + wave;
    int ti = tile >> 8;          // 4096/16 = 256 tiles per dim
    int tj = tile & 255;
    int half = lane >> 4, lm = lane & 15;
    // A layout (16x4 f32): lanes 0-15 hold M=lm, K={k0,k0+1}; lanes 16-31 K={k0+2,k0+3}
    const float* arow = raw + (size_t)(ti * 16 + lm) * 64 + 2 * half;
    const float* brow = raw + (size_t)(tj * 16 + lm) * 64 + 2 * half;
    v8f c = {0.f, 0.f, 0.f, 0.f, 0.f, 0.f, 0.f, 0.f};
#pragma unroll
    for (int k0 = 0; k0 < 64; k0 += 4) {
        v2f a, b;
        a.x = arow[k0]; a.y = arow[k0 + 1];
        b.x = brow[k0]; b.y = brow[k0 + 1];
        c = __builtin_amdgcn_wmma_f32_16x16x4_f32(false, a, false, b,
                                                  (short)0, c, false, false);
    }
    // D layout: vgpr r -> M = r + half*8, N = lm
    int gj = tj * 16 + lm;
    float sj = sq[gj];
#pragma unroll
    for (int r = 0; r < 8; ++r) {
        int gi = ti * 16 + r + half * 8;
        d2[(size_t)gi * B_N + gj] = sq[gi] + sj - 2.f * c[r];
    }
}

// ---------- 25 nearest neighbors per row ----------
// Single global pass: each thread caches its 16 strided elements in registers,
// then 25 lexicographic (val,idx) min-selection rounds run register-resident.
__global__ void k_topk(float* fs) {
    const float* d2 = fs + WS_D2;
    int* nbrs = (int*)(fs + WS_NBR);
    __shared__ float sv[256];
    __shared__ int si[256];
    __shared__ float lastV;
    __shared__ int lastI;
    int i = blockIdx.x, tid = threadIdx.x;
    const float* row = d2 + (size_t)i * B_N;
    float val[16];
#pragma unroll
    for (int c = 0; c < 16; ++c) val[c] = row[tid + (c << 8)];
    if (tid == 0) { lastV = -3.4e38f; lastI = -1; }
    __syncthreads();
    for (int sel = 0; sel < K_NB; ++sel) {
        float lv = lastV; int li = lastI;
        float bv = 3.4e38f; int bi = 0x7fffffff;
#pragma unroll
        for (int c = 0; c < 16; ++c) {
            int j = tid + (c << 8);
            float v = val[c];
            bool elig = (j != i) && ((v > lv) || (v == lv && j > li));
            if (elig && (v < bv || (v == bv && j < bi))) { bv = v; bi = j; }
        }
        sv[tid] = bv; si[tid] = bi; __syncthreads();
        for (int s = 128; s > 0; s >>= 1) {
            if (tid < s) {
                float ov = sv[tid + s]; int oi = si[tid + s];
                if (ov < sv[tid] || (ov == sv[tid] && oi < si[tid])) { sv[tid] = ov; si[tid] = oi; }
            }
            __syncthreads();
        }
        if (tid == 0) { nbrs[i * K_NB + sel] = si[0]; lastV = sv[0]; lastI = si[0]; }
        __syncthreads();
    }
}

// ---------- TSA: top eigenvector of each 25-NN covariance (latent & raw), ----------
// ---------- accumulate 2 - 2*(u.v)^2. ONE WAVE (32 threads) per point: ----------
// ---------- single-wave workgroup => barriers are S_NOP (free on CDNA5). ----------
__global__ void k_tsa(const float* __restrict__ latent, const float* __restrict__ raw,
                      float* fs) {
    __shared__ float Y[K_NB][64];
    __shared__ float v[64], w[K_NB], red[32], u[64];
    __shared__ int nb[K_NB];
    int b = blockIdx.x, t = threadIdx.x;   // t in [0,32); lane owns columns t and t+32
    int t2 = t + 32;
    const int* nbrs = (const int*)(fs + WS_NBR);
    if (t < K_NB) nb[t] = nbrs[b * K_NB + t];
    __syncthreads();
    for (int pass = 0; pass < 2; ++pass) {
        const float* src = (pass == 0) ? latent : raw;
#pragma unroll
        for (int r = 0; r < K_NB; ++r) {
            Y[r][t]  = src[(size_t)nb[r] * 64 + t];
            Y[r][t2] = src[(size_t)nb[r] * 64 + t2];
        }
        // center over the K=25 neighbors (columns private to this lane)
        float m0 = 0.f, m1 = 0.f;
#pragma unroll
        for (int r = 0; r < K_NB; ++r) { m0 += Y[r][t]; m1 += Y[r][t2]; }
        m0 *= (1.f / (float)K_NB); m1 *= (1.f / (float)K_NB);
#pragma unroll
        for (int r = 0; r < K_NB; ++r) { Y[r][t] -= m0; Y[r][t2] -= m1; }
        v[t] = 1.f + 0.01f * (float)t;
        v[t2] = 1.f + 0.01f * (float)t2;
        __syncthreads();
        for (int it = 0; it < 40; ++it) {
            if (t < K_NB) {
                float s = 0.f;
#pragma unroll
                for (int d = 0; d < 64; ++d) s += Y[t][d] * v[d];
                w[t] = s;
            }
            __syncthreads();
            float s0 = 0.f, s1 = 0.f;
#pragma unroll
            for (int r = 0; r < K_NB; ++r) { s0 += Y[r][t] * w[r]; s1 += Y[r][t2] * w[r]; }
            red[t] = s0 * s0 + s1 * s1; __syncthreads();
            for (int st = 16; st > 0; st >>= 1) { if (t < st) red[t] += red[t + st]; __syncthreads(); }
            float nr = rsqrtf(red[0] + 1e-30f);
            __syncthreads();
            v[t] = s0 * nr; v[t2] = s1 * nr; __syncthreads();
        }
        if (pass == 0) { u[t] = v[t]; u[t2] = v[t2]; __syncthreads(); }
    }
    red[t] = u[t] * v[t] + u[t2] * v[t2]; __syncthreads();
    for (int st = 16; st > 0; st >>= 1) { if (t < st) red[t] += red[t + st]; __syncthreads(); }
    if (t == 0) {
        float dt = red[0];
        atomicAdd(&fs[WS_TSA], 2.f - 2.f * dt * dt);
    }
}

// ---------- combine everything ----------
__global__ void k_final(const float* fs, float* out) {
    float recon = fs[WS_RECON] / (float)(B_N * D_N);
    float tr = fs[WS_TRACE], fb = fs[WS_FROB], lm = fs[WS_LMAX];
    float pr = 0.01f * (tr * tr / fb);
    float an = 0.01f * (1.f - lm / tr);
    float tsa = 0.1f * (fs[WS_TSA] / (float)B_N);
    out[0] = recon + pr + an + tsa;
}

extern "C" void kernel_launch(void* const* d_in, const int* in_sizes, int n_in,
                              void* d_out, int out_size, void* d_ws, size_t ws_size,
                              hipStream_t stream) {
    (void)in_sizes; (void)n_in; (void)out_size; (void)ws_size;
    const float* outputs = (const float*)d_in[0];
    const float* targets = (const float*)d_in[1];
    const float* latent  = (const float*)d_in[2];
    const float* raw     = (const float*)d_in[3];
    float* fs = (float*)d_ws;
    float* out = (float*)d_out;

    k_init   <<<1, 32, 0, stream>>>(fs);
    k_recon  <<<256, 256, 0, stream>>>(outputs, targets, fs);
    k_colmean<<<64, 256, 0, stream>>>(latent, fs);
    k_cov    <<<64, 256, 0, stream>>>(latent, fs);
    k_eig    <<<1, 64, 0, stream>>>(fs);
    k_sq     <<<16, 256, 0, stream>>>(raw, fs);
    k_gram   <<<8192, 256, 0, stream>>>(raw, fs);   // 65536 16x16 tiles / 8 waves
    k_topk   <<<4096, 256, 0, stream>>>(fs);
    k_tsa    <<<4096, 32, 0, stream>>>(latent, raw, fs);  // 1 wave/block: free barriers
    k_final  <<<1, 1, 0, stream>>>(fs, out);
}